// ConditionDenseCapsule_51754355917250
// MI455X (gfx1250) — compile-verified
//
#include <hip/hip_runtime.h>
#include <math.h>

// ---------------------------------------------------------------------------
// Matrix-capsule EM routing, MI455X (gfx1250, wave32).
//
// votes[b, n=t*CH+c, j, d=p*4+r] = sum_q w[c,j,p,q] * pose[b,t,c,q*4+r]
// mapped to V_WMMA_F32_16X16X4_F32:  A = w[c] rows (j,p) x K=q,
//                                    B = pose cols (t,r) x K=q,
//                                    D = 16x16 vote tile.
// votes are regenerated each EM iteration into LDS (never hit HBM): inputs
// (pose 16.8MB + act 1MB + w 64KB) are L2-resident, so 3 fused passes replace
// ~4.3GB of materialized-votes traffic.
// ---------------------------------------------------------------------------

typedef __attribute__((ext_vector_type(2))) float v2f;
typedef __attribute__((ext_vector_type(8))) float v8f;

#define B_      2
#define NI_     4096
#define CH_     32
#define J_      32
#define D2_     16
#define N_ITER_ 3
#define EPS_    1e-6f

#define TT        32                     // t-values (n's per fixed c) per workgroup
#define NT_CHUNKS (NI_ / TT)             // 128
#define NWG       (B_ * CH_ * NT_CHUNKS) // 8192
#define COMPS     33                     // rsum + m1[16] + m2[16]  (also mu,invsig,const)
#define PSTRIDE   (J_ * COMPS)           // 1056 floats per workgroup partial

// padded LDS vote tile strides (bank-conflict-free for scatter and j-reads)
#define VS_D 33
#define VS_T (16 * VS_D)                 // 528

// ---------------------------------------------------------------------------
// Pass kernel: one workgroup = (b, c, chunk of 32 t's). 256 threads = 8 waves.
// Stage A: WMMA-generate the 32x32x16 vote tile into LDS.
// Stage B: per-n E-step (on-the-fly r) + moment accumulation in registers.
// Stage C: cross-wave LDS reduce, write one [J][33] partial per workgroup.
// ---------------------------------------------------------------------------
__global__ __launch_bounds__(256)
void em_pass_kernel(const float* __restrict__ pose,
                    const float* __restrict__ act,
                    const float* __restrict__ w,
                    const float* __restrict__ stats,  // [B][J][33]: mu16, invsig16, const
                    float* __restrict__ partials,
                    int iter)
{
    extern __shared__ float smem[];
    float* votes_s = smem;                  // TT * VS_T = 16896 floats
    float* pose_s  = votes_s + TT * VS_T;   // 512
    float* w_s     = pose_s + TT * 16;      // 512
    float* stats_s = w_s + 512;             // 1056

    const int bx   = blockIdx.x;
    const int tc   = bx % NT_CHUNKS;
    const int c    = (bx / NT_CHUNKS) % CH_;
    const int b    = bx / (NT_CHUNKS * CH_);
    const int t0   = tc * TT;
    const int tid  = threadIdx.x;
    const int lane = tid & 31;
    const int wv   = tid >> 5;

    // ---- cooperative staging ----
    for (int idx = tid; idx < TT * 16; idx += 256) {
        int t = idx >> 4, d = idx & 15;
        pose_s[idx] = pose[(((size_t)b * NI_ + (t0 + t)) * CH_ + c) * 16 + d];
    }
    for (int idx = tid; idx < 512; idx += 256)
        w_s[idx] = w[(size_t)c * 512 + idx];
    if (iter > 0) {
        for (int idx = tid; idx < J_ * COMPS; idx += 256)
            stats_s[idx] = stats[(size_t)b * J_ * COMPS + idx];
    }
    __syncthreads();

    // ---- Stage A: WMMA vote generation. Wave wv owns M-tile wv (rows j*4+p).
    {
        const int rowA = lane & 15;
        const int R    = wv * 16 + rowA;          // global A row
        const int jA   = R >> 2, pA = R & 3;
        const int qb   = (lane < 16) ? 0 : 2;     // K split per ISA A/B layout
        v2f a;
        a.x = w_s[jA * 16 + pA * 4 + qb];         // A[row][qb]
        a.y = w_s[jA * 16 + pA * 4 + qb + 1];     // A[row][qb+1]

        const int colB = lane & 15;
        const int rbase = (lane < 16) ? 0 : 8;
#pragma unroll
        for (int nt = 0; nt < (TT * 4) / 16; ++nt) {  // 8 N-tiles of 16 cols
            const int gcol = nt * 16 + colB;          // global col = t_local*4 + r
            const int tl = gcol >> 2, r = gcol & 3;
            v2f bm;
            bm.x = pose_s[tl * 16 + qb * 4 + r];          // B[qb][col]
            bm.y = pose_s[tl * 16 + (qb + 1) * 4 + r];    // B[qb+1][col]
            v8f acc = {0.f, 0.f, 0.f, 0.f, 0.f, 0.f, 0.f, 0.f};
            acc = __builtin_amdgcn_wmma_f32_16x16x4_f32(
                false, a, false, bm, (short)0, acc, false, false);
            // D layout: VGPR e -> row e+rbase (this lane), col = gcol
#pragma unroll
            for (int e = 0; e < 8; ++e) {
                const int Rr = wv * 16 + rbase + e;
                const int j = Rr >> 2, p = Rr & 3;
                votes_s[tl * VS_T + (p * 4 + r) * VS_D + j] = acc[e];
            }
        }
    }
    __syncthreads();

    // ---- Stage B: EM accumulation. Lane = capsule j; wave handles 4 n's.
    float rsum = 0.f;
    float m1[16], m2[16];
#pragma unroll
    for (int i = 0; i < 16; ++i) { m1[i] = 0.f; m2[i] = 0.f; }

    for (int k = 0; k < TT / 8; ++k) {
        const int t = wv + k * 8;
        float vd[16];
#pragma unroll
        for (int i = 0; i < 16; ++i)
            vd[i] = votes_s[t * VS_T + i * VS_D + lane];

        float rj;
        if (iter == 0) {
            rj = 1.0f / (float)J_;
        } else {
            // logits_j = [log(a_j+eps) - 0.5*sum log(sig2)] - 0.5*sum diff2/sig2
            const float* st = stats_s + lane * COMPS;
            float lg = st[32];
#pragma unroll
            for (int i = 0; i < 16; ++i) {
                const float dlt = vd[i] - st[i];
                lg -= 0.5f * dlt * dlt * st[16 + i];
            }
            // softmax across the 32 j-lanes (wave32 shuffle reduction)
            float mx = lg;
#pragma unroll
            for (int off = 16; off >= 1; off >>= 1)
                mx = fmaxf(mx, __shfl_xor(mx, off, 32));
            const float ex = __expf(lg - mx);
            float sm = ex;
#pragma unroll
            for (int off = 16; off >= 1; off >>= 1)
                sm += __shfl_xor(sm, off, 32);
            rj = ex / sm;
        }

        const float av = act[((size_t)b * NI_ + (t0 + t)) * CH_ + c];
        const float ra = rj * av;
        rsum += ra;
#pragma unroll
        for (int i = 0; i < 16; ++i) {
            m1[i] += ra * vd[i];
            m2[i] += ra * vd[i] * vd[i];
        }
    }
    __syncthreads();

    // ---- Stage C: cross-wave reduce (reuse vote region), one partial per WG.
    float* red = smem;                          // [8][32][33] = 8448 < 16896
    red[(wv * 32 + lane) * COMPS + 0] = rsum;
#pragma unroll
    for (int i = 0; i < 16; ++i) {
        red[(wv * 32 + lane) * COMPS + 1 + i]  = m1[i];
        red[(wv * 32 + lane) * COMPS + 17 + i] = m2[i];
    }
    __syncthreads();
    for (int idx = tid; idx < PSTRIDE; idx += 256) {
        const int j = idx / COMPS, comp = idx % COMPS;
        float s = 0.f;
        for (int ww = 0; ww < 8; ++ww)
            s += red[(ww * 32 + j) * COMPS + comp];
        partials[(size_t)bx * PSTRIDE + idx] = s;
    }
}

// ---------------------------------------------------------------------------
// Deterministic second-level reduce: sums[b][j][comp] over 4096 partials.
// Fully coalesced (adjacent threads read adjacent floats each step).
// ---------------------------------------------------------------------------
__global__ __launch_bounds__(256)
void em_reduce_kernel(const float* __restrict__ partials, float* __restrict__ sums)
{
    const int idx = blockIdx.x * 256 + threadIdx.x;
    if (idx >= B_ * PSTRIDE) return;
    const int b = idx / PSTRIDE;
    const int rem = idx % PSTRIDE;
    const int G = CH_ * NT_CHUNKS;   // 4096 workgroups per batch
    const float* p = partials + (size_t)b * G * PSTRIDE + rem;
    float s = 0.f;
    for (int g = 0; g < G; ++g) s += p[(size_t)g * PSTRIDE];
    sums[idx] = s;
}

// ---------------------------------------------------------------------------
// Finalize: mu, sigma2 (exact eps-corrected moment form), cost, a_j,
// next-iteration E-step stats; last iteration writes both output tuples.
// ---------------------------------------------------------------------------
__global__ void em_finalize_kernel(const float* __restrict__ sums,
                                   const float* __restrict__ beta_v,
                                   const float* __restrict__ beta_a,
                                   float* __restrict__ stats,
                                   float* __restrict__ out,
                                   int iter)
{
    const int t = threadIdx.x;
    if (t >= B_ * J_) return;
    const int b = t / J_, j = t % J_;
    const float* s = sums + (size_t)t * COMPS;

    const float rsum  = s[0];
    const float den   = rsum + EPS_;
    const float alpha = rsum / den;                // sum of coeff over n
    const float lam   = 0.01f * (1.0f - powf(0.95f, (float)(iter + 1)));

    float mu[16];
    float cost = 0.f, slog = 0.f, invsg[16];
#pragma unroll
    for (int i = 0; i < 16; ++i) {
        mu[i] = s[1 + i] / den;
        const float ev2 = s[17 + i] / den;
        float var = ev2 - (2.0f - alpha) * mu[i] * mu[i];   // == sum coeff*(v-mu)^2
        var = fmaxf(var, 0.0f) + EPS_;
        const float lg = __logf(var);
        slog += lg;
        invsg[i] = 1.0f / var;
        cost += (beta_v[j * 16 + i] + 0.5f * lg) * rsum;
    }
    const float aj = 1.0f / (1.0f + __expf(-(lam * (beta_a[j] - cost))));

    float* st = stats + (size_t)t * COMPS;
#pragma unroll
    for (int i = 0; i < 16; ++i) { st[i] = mu[i]; st[16 + i] = invsg[i]; }
    st[32] = __logf(aj + EPS_) - 0.5f * slog;

    if (iter == N_ITER_ - 1) {
        const size_t o1 = ((size_t)b * J_ + j) * 17;
        const size_t o2 = (size_t)B_ * J_ * 17 + o1;   // capsules2 == capsules1
#pragma unroll
        for (int i = 0; i < 16; ++i) { out[o1 + i] = mu[i]; out[o2 + i] = mu[i]; }
        out[o1 + 16] = aj;
        out[o2 + 16] = aj;
    }
}

// ---------------------------------------------------------------------------
extern "C" void kernel_launch(void* const* d_in, const int* in_sizes, int n_in,
                              void* d_out, int out_size, void* d_ws, size_t ws_size,
                              hipStream_t stream)
{
    (void)in_sizes; (void)n_in; (void)out_size; (void)ws_size;
    const float* pose   = (const float*)d_in[0];
    const float* act    = (const float*)d_in[1];
    const float* w      = (const float*)d_in[2];
    const float* beta_v = (const float*)d_in[3];
    const float* beta_a = (const float*)d_in[4];
    float* out = (float*)d_out;

    float* partials = (float*)d_ws;                            // 8192*1056 floats (~34.6MB)
    float* sums     = partials + (size_t)NWG * PSTRIDE;        // 2112 floats
    float* stats    = sums + (size_t)B_ * PSTRIDE;             // 2112 floats

    const size_t shmem =
        (size_t)(TT * VS_T + TT * 16 + 512 + J_ * COMPS) * sizeof(float); // ~76KB

    for (int iter = 0; iter < N_ITER_; ++iter) {
        em_pass_kernel<<<NWG, 256, shmem, stream>>>(pose, act, w, stats, partials, iter);
        em_reduce_kernel<<<(B_ * PSTRIDE + 255) / 256, 256, 0, stream>>>(partials, sums);
        em_finalize_kernel<<<1, 64, 0, stream>>>(sums, beta_v, beta_a, stats, out, iter);
    }
}